// RandomDropout_7842610283498
// MI455X (gfx1250) — compile-verified
//
#include <hip/hip_runtime.h>
#include <hip/hip_bf16.h>
#include <stdint.h>

// ---------------------------------------------------------------------------
// RandomDropout + compact:  B=8192 rows of S=2048 int32 tokens.
// out[row][j] = (row even || ntok<10) ? in[row][j]
//             : (j < d ? in[j] : (j+1<S ? in[j+1] : 0))
// where d = argmin over s<ntok of JAX threefry-uniform(key=42) for flat index
// row*S+s, ties -> lowest s (identical to top_k(-r,1)).
// Memory-bound: 128 MiB @ 23.3 TB/s ~ 5.5us floor. Threefry only on odd rows'
// valid prefix, overlapped with the stream.
// ---------------------------------------------------------------------------

#define S_LEN 2048

// ---- gfx1250 async global->LDS staging (guarded; fallback = plain loads) ----
#if defined(__HIP_DEVICE_COMPILE__) && defined(__gfx1250__) && \
    __has_builtin(__builtin_amdgcn_global_load_async_to_lds_b128) && \
    __has_builtin(__builtin_amdgcn_s_wait_asynccnt)
#define HAVE_ASYNC_LDS 1
#else
#define HAVE_ASYNC_LDS 0
#endif

#if HAVE_ASYNC_LDS
// Builtin expects pointers to int __attribute__((vector_size(16))) per the
// clang diagnostic; spell the vector type exactly that way.
typedef int v4i_t __attribute__((vector_size(4 * sizeof(int))));
typedef __attribute__((address_space(1))) v4i_t g_v4i;   // global
typedef __attribute__((address_space(3))) v4i_t l_v4i;   // LDS
// Global generic addresses are identity in AS1; LDS offset = low 32 bits of a
// generic shared pointer (aperture scheme) -> build AS pointers via integer
// casts, avoiding addrspacecast legality questions.
#define GLB_PTR(p) ((g_v4i*)(uintptr_t)(p))
#define LDS_PTR(p) ((l_v4i*)(uint32_t)(uintptr_t)(p))
#endif

__device__ __forceinline__ unsigned rotl32(unsigned x, unsigned n) {
    return (x << n) | (x >> (32u - n));
}

// Threefry-2x32, 20 rounds, key = (0, 42)  == jax.random.key(42)
__device__ __forceinline__ void threefry2x32_key42(unsigned c0, unsigned c1,
                                                   unsigned& o0, unsigned& o1) {
    const unsigned ks0 = 0u;
    const unsigned ks1 = 42u;
    const unsigned ks2 = 0x1BD11BDAu ^ 42u;  // k0 ^ k1 ^ parity
    unsigned x0 = c0 + ks0;
    unsigned x1 = c1 + ks1;
#define TF_R(r) { x0 += x1; x1 = rotl32(x1, (r)); x1 ^= x0; }
    TF_R(13) TF_R(15) TF_R(26) TF_R(6)   x0 += ks1; x1 += ks2 + 1u;
    TF_R(17) TF_R(29) TF_R(16) TF_R(24)  x0 += ks2; x1 += ks0 + 2u;
    TF_R(13) TF_R(15) TF_R(26) TF_R(6)   x0 += ks0; x1 += ks1 + 3u;
    TF_R(17) TF_R(29) TF_R(16) TF_R(24)  x0 += ks1; x1 += ks2 + 4u;
    TF_R(13) TF_R(15) TF_R(26) TF_R(6)   x0 += ks2; x1 += ks0 + 5u;
#undef TF_R
    o0 = x0; o1 = x1;
}

__global__ __launch_bounds__(256)
void RandomDropout_7842610283498_kernel(const int* __restrict__ in,
                                        int* __restrict__ out,
                                        unsigned rowHalf,   // half / S
                                        unsigned half) {    // (B*S)/2
    const unsigned row = blockIdx.x;
    const unsigned t   = threadIdx.x;
    const int* inRow = in  + (size_t)row * S_LEN;
    int*      outRow = out + (size_t)row * S_LEN;

    // Even rows: apply_row is false -> straight copy, skip everything else.
    if ((row & 1u) == 0u) {
        int4 v0 = ((const int4*)inRow)[t];
        int4 v1 = ((const int4*)inRow)[t + 256];
        ((int4*)outRow)[t]       = v0;
        ((int4*)outRow)[t + 256] = v1;
        return;
    }

    __shared__ int tile[S_LEN];
    __shared__ unsigned s_fz;                 // first zero index (= n_tokens)
    __shared__ unsigned long long s_key;      // min ((bits>>9)<<32 | s)

    if (t == 0u) { s_fz = S_LEN; s_key = ~0ull; }

    // ---- Stage the 8 KB row into LDS (async on gfx1250) ----
#if HAVE_ASYNC_LDS
    {
        const char* g = (const char*)inRow;
        char*       l = (char*)tile;
        __builtin_amdgcn_global_load_async_to_lds_b128(
            GLB_PTR(g + 16u * t),          LDS_PTR(l + 16u * t),          0, 0);
        __builtin_amdgcn_global_load_async_to_lds_b128(
            GLB_PTR(g + 16u * (t + 256u)), LDS_PTR(l + 16u * (t + 256u)), 0, 0);
        __builtin_amdgcn_s_wait_asynccnt(0);
    }
#else
    ((int4*)tile)[t]       = ((const int4*)inRow)[t];
    ((int4*)tile)[t + 256] = ((const int4*)inRow)[t + 256];
#endif
    __syncthreads();

    // ---- n_tokens = index of first zero (contiguous positive prefix) ----
    unsigned fz = S_LEN;
#pragma unroll
    for (int k = 0; k < 8; ++k) {
        unsigned j = t + 256u * (unsigned)k;
        if (tile[j] == 0 && j < fz) fz = j;
    }
#pragma unroll
    for (int off = 16; off > 0; off >>= 1) {
        unsigned o = (unsigned)__shfl_xor((int)fz, off, 32);
        if (o < fz) fz = o;
    }
    if ((t & 31u) == 0u) atomicMin(&s_fz, fz);
    __syncthreads();

    const unsigned ntok  = s_fz;
    const bool     apply = (ntok >= 10u);     // row is odd here

    // ---- Threefry argmin over the valid prefix (JAX-exact) ----
    unsigned long long lk = ~0ull;
    if (apply) {
        // flat g = row*S + s ; lane0 block = g, lane1 block = g - half
        const bool     lane1 = (row >= rowHalf);
        const unsigned base  = lane1 ? (row * (unsigned)S_LEN - half)
                                     : (row * (unsigned)S_LEN);
#pragma unroll
        for (int k = 0; k < 8; ++k) {
            unsigned j = t + 256u * (unsigned)k;
            if (j < ntok) {
                unsigned o0, o1;
                threefry2x32_key42(base + j, base + j + half, o0, o1);
                unsigned bits = lane1 ? o1 : o0;
                unsigned long long key =
                    (((unsigned long long)(bits >> 9)) << 32) | (unsigned long long)j;
                if (key < lk) lk = key;
            }
        }
    }
    {
        // 64-bit butterfly min via two 32-bit shuffles (wave32)
        unsigned lo = (unsigned)lk, hi = (unsigned)(lk >> 32);
#pragma unroll
        for (int off = 16; off > 0; off >>= 1) {
            unsigned olo = (unsigned)__shfl_xor((int)lo, off, 32);
            unsigned ohi = (unsigned)__shfl_xor((int)hi, off, 32);
            unsigned long long o = (((unsigned long long)ohi) << 32) | olo;
            if (o < lk) { lk = o; lo = olo; hi = ohi; }
        }
    }
    if ((t & 31u) == 0u) atomicMin(&s_key, lk);
    __syncthreads();

    const unsigned d = apply ? (unsigned)(s_key & 0xffffffffu) : (unsigned)S_LEN;

    // ---- Shift-by-one compaction from LDS, vectorized b128 stores ----
    const unsigned j0 = t * 8u;
    int a[9];
    *(int4*)&a[0] = *(const int4*)&tile[j0];
    *(int4*)&a[4] = *(const int4*)&tile[j0 + 4u];
    a[8] = (j0 + 8u < (unsigned)S_LEN) ? tile[j0 + 8u] : 0;

    int o[8];
#pragma unroll
    for (int k = 0; k < 8; ++k)
        o[k] = ((j0 + (unsigned)k) < d) ? a[k] : a[k + 1];

    *(int4*)&outRow[j0]      = *(const int4*)&o[0];
    *(int4*)&outRow[j0 + 4u] = *(const int4*)&o[4];
}

extern "C" void kernel_launch(void* const* d_in, const int* in_sizes, int n_in,
                              void* d_out, int out_size, void* d_ws, size_t ws_size,
                              hipStream_t stream) {
    (void)n_in; (void)out_size; (void)d_ws; (void)ws_size;
    const int* in  = (const int*)d_in[0];
    int*       out = (int*)d_out;

    const unsigned total   = (unsigned)in_sizes[0];     // B * S
    const unsigned B       = total / (unsigned)S_LEN;   // 8192
    const unsigned half    = total / 2u;                // iota split point
    const unsigned rowHalf = half / (unsigned)S_LEN;    // first lane-1 row

    dim3 grid(B), block(256);
    RandomDropout_7842610283498_kernel<<<grid, block, 0, stream>>>(
        in, out, rowHalf, half);
}